// DistGAT_13348758356518
// MI455X (gfx1250) — compile-verified
//
#include <hip/hip_runtime.h>
#include <math.h>

// ---------------------------------------------------------------------------
// Problem constants (from the reference)
// ---------------------------------------------------------------------------
#define N_NODES   50000
#define N_EDGES   800000
#define F_IN      256
#define HID       64
#define H0        4
#define NCLS      40
#define F0        (H0 * HID)     // 256 = layer-0 output width (= layer-1 input)
#define NEG_SLOPE 0.2f

typedef float v2f __attribute__((ext_vector_type(2)));
typedef float v8f __attribute__((ext_vector_type(8)));
typedef unsigned int u32x4 __attribute__((ext_vector_type(4)));
typedef int i32x4 __attribute__((ext_vector_type(4)));
typedef int i32x8 __attribute__((ext_vector_type(8)));

// ---------------------------------------------------------------------------
// Order-preserving float <-> uint encoding for atomic max on floats
// ---------------------------------------------------------------------------
__device__ __forceinline__ unsigned int f2ord(float f) {
  unsigned int b = __float_as_uint(f);
  return (b & 0x80000000u) ? ~b : (b | 0x80000000u);
}
__device__ __forceinline__ float ord2f(unsigned int k) {
  unsigned int b = (k & 0x80000000u) ? (k ^ 0x80000000u) : ~k;
  return __uint_as_float(b);
}

// ---------------------------------------------------------------------------
// Z[N x ncol] = X[N x 256] * W[256 x ncol], fp32, via V_WMMA_F32_16X16X4_F32.
//   grid.x = N/16 (N divisible by 16), grid.y covers ncol in 128-col slabs.
//   block = 256 threads = 8 waves; A-tile (16x256 fp32, 16 KB) staged in LDS
//   by the Tensor Data Mover (2-D tile DMA, TENSORcnt) when available.
// A fragment (16x4):  lanes 0-15 hold K=k+0,k+1; lanes 16-31 hold K=k+2,k+3.
// B fragment (4x16):  lane n<16 -> cols, VGPR0/1 = rows k+0/k+1 (hi half k+2/3).
// C/D (16x16 f32):    lane n<16 -> N=n, VGPR v -> M=v (hi half M=8+v).
// ---------------------------------------------------------------------------
__global__ void gemm_wmma_f32(const float* __restrict__ X,
                              const float* __restrict__ W,
                              float* __restrict__ Z,
                              int ncol) {
  __shared__ float ldsA[16 * 256];

  const int m0 = blockIdx.x * 16;

#if defined(__has_builtin) && __has_builtin(__builtin_amdgcn_tensor_load_to_lds)
  // --- TDM tile load: 16 rows x 256 fp32 from X[m0..m0+15][*] into LDS. ---
  if (threadIdx.x < 32) {   // wave 0 issues the DMA once (TDM ignores EXEC)
    unsigned long long ga =
        (unsigned long long)(uintptr_t)(X + (size_t)m0 * 256);
    unsigned int ldsoff = (unsigned int)(uintptr_t)&ldsA[0]; // addr[31:0] = LDS offset
    u32x4 g0;
    g0.x = 1u;                                          // count=1 valid user D#
    g0.y = ldsoff;                                      // lds_addr (bytes)
    g0.z = (unsigned int)(ga & 0xFFFFFFFFu);            // global_addr[31:0]
    g0.w = (unsigned int)((ga >> 32) & 0x01FFFFFFu)     // global_addr[56:32]
           | (2u << 30);                                // type = 2 ("image")
    i32x8 g1;
    g1[0] = (2 << 16);                 // workgroup_mask=0, data_size=2 (4 bytes)
    g1[1] = (256 & 0xFFFF) << 16;      // tensor_dim0[15:0] = 256
    g1[2] = (16 & 0xFFFF) << 16;       // tensor_dim0[31:16]=0 | tensor_dim1[15:0]=16
    g1[3] = (256 << 16);               // tensor_dim1[31:16]=0 | tile_dim0 = 256
    g1[4] = 16;                        // tile_dim1 = 16, tile_dim2 = 0
    g1[5] = 256;                       // tensor_dim0_stride[31:0] = 256
    g1[6] = 0;                         // stride0[47:32]=0 | stride1[15:0]=0
    g1[7] = 0;                         // stride1[47:16] = 0
    i32x4 gz4 = {0, 0, 0, 0};          // groups 2/3 unused (2-D tile)
    i32x8 gz8 = {0, 0, 0, 0, 0, 0, 0, 0};
    __builtin_amdgcn_tensor_load_to_lds(g0, g1, gz4, gz4, gz8, 0);
    __builtin_amdgcn_s_wait_tensorcnt(0);
  }
  __syncthreads();
#else
  // --- Fallback: cooperative, coalesced float4 staging. ---
  {
    const int t  = threadIdx.x;          // 0..255
    const int r  = t >> 4;               // row 0..15
    const int c0 = (t & 15) * 16;        // 16 consecutive floats
    const float4* s = (const float4*)(X + (size_t)(m0 + r) * F0 + c0);
    float4*       d = (float4*)(ldsA + r * 256 + c0);
    d[0] = s[0]; d[1] = s[1]; d[2] = s[2]; d[3] = s[3];
  }
  __syncthreads();
#endif

  const int wave = threadIdx.x >> 5;     // 0..7
  const int lane = threadIdx.x & 31;
  const int n0   = blockIdx.y * 128 + wave * 16;
  if (n0 >= ncol) return;                // wave-uniform: EXEC stays all-ones

  const int lrow  = lane & 15;
  const int khalf = (lane >> 4) * 2;     // 0 or 2

  // Clamp B column for the ragged ncol=40 tile (those D columns are never stored).
  int bcol = n0 + lrow;
  if (bcol >= ncol) bcol = ncol - 1;

  v8f acc = {};
  for (int k = 0; k < 256; k += 4) {
    v2f a, b;
    a.x = ldsA[lrow * 256 + k + khalf];
    a.y = ldsA[lrow * 256 + k + khalf + 1];
    b.x = W[(size_t)(k + khalf) * ncol + bcol];
    b.y = W[(size_t)(k + khalf + 1) * ncol + bcol];
    acc = __builtin_amdgcn_wmma_f32_16x16x4_f32(false, a, false, b,
                                                (short)0, acc, false, false);
  }

  const int col = n0 + lrow;
  if (col < ncol) {
    const int rbase = m0 + (lane >> 4) * 8;
#pragma unroll
    for (int v = 0; v < 8; ++v)
      Z[(size_t)(rbase + v) * ncol + col] = acc[v];
  }
}

// ---------------------------------------------------------------------------
// el[n,h] = dot(z[n,h,:], attn_l[h,:]);  er likewise.  grid=N, block=32*H.
// ---------------------------------------------------------------------------
__global__ void el_er_kernel(const float* __restrict__ Z,
                             const float* __restrict__ al,
                             const float* __restrict__ ar,
                             float* __restrict__ el, float* __restrict__ er,
                             int H, int D) {
  const int n    = blockIdx.x;
  const int h    = threadIdx.x >> 5;
  const int lane = threadIdx.x & 31;
  float sl = 0.f, sr = 0.f;
  for (int d = lane; d < D; d += 32) {
    float z = Z[((size_t)n * H + h) * D + d];
    sl += z * al[h * D + d];
    sr += z * ar[h * D + d];
  }
#pragma unroll
  for (int off = 16; off; off >>= 1) {
    sl += __shfl_down(sl, off, 32);
    sr += __shfl_down(sr, off, 32);
  }
  if (lane == 0) { el[(size_t)n * H + h] = sl; er[(size_t)n * H + h] = sr; }
}

// ---------------------------------------------------------------------------
// Init helpers (re-run every call: deterministic under graph replay).
// ---------------------------------------------------------------------------
__global__ void init_maxsum(unsigned int* __restrict__ m_u,
                            float* __restrict__ s, int count) {
  int i = blockIdx.x * blockDim.x + threadIdx.x;
  if (i < count) { m_u[i] = 0u; s[i] = 0.f; }   // ord-0 < ord(any finite)
}
__global__ void init_out_bias(float* __restrict__ out,
                              const float* __restrict__ bias,
                              int n, int fdim) {
  int i = blockIdx.x * blockDim.x + threadIdx.x;
  if (i < n * fdim) out[i] = bias[i % fdim];
}

// ---------------------------------------------------------------------------
// Pass A: e = leaky_relu(el[src]+er[dst]); store; atomic segment-max on dst.
// ---------------------------------------------------------------------------
__global__ void edge_logits(const int* __restrict__ src,
                            const int* __restrict__ dst,
                            const float* __restrict__ el,
                            const float* __restrict__ er,
                            float* __restrict__ ebuf,
                            unsigned int* __restrict__ m_u, int H) {
  int t = blockIdx.x * blockDim.x + threadIdx.x;
  if (t >= N_EDGES * H) return;
  int e = t / H, h = t - e * H;
  float v = el[(size_t)src[e] * H + h] + er[(size_t)dst[e] * H + h];
  v = v > 0.f ? v : NEG_SLOPE * v;
  ebuf[t] = v;
  atomicMax(&m_u[(size_t)dst[e] * H + h], f2ord(v));
}

// ---------------------------------------------------------------------------
// Pass B: a = exp(e - max[dst]); overwrite ebuf; atomic segment-sum on dst.
// ---------------------------------------------------------------------------
__global__ void edge_exp(const int* __restrict__ dst,
                         float* __restrict__ ebuf,
                         const unsigned int* __restrict__ m_u,
                         float* __restrict__ s, int H) {
  int t = blockIdx.x * blockDim.x + threadIdx.x;
  if (t >= N_EDGES * H) return;
  int e = t / H, h = t - e * H;
  float a = __expf(ebuf[t] - ord2f(m_u[(size_t)dst[e] * H + h]));
  ebuf[t] = a;
  atomicAdd(&s[(size_t)dst[e] * H + h], a);
}

// ---------------------------------------------------------------------------
// Pass C: out[dst] += (a / s[dst]) * z[src].  One wave per edge; lane-strided
// feature loop -> coalesced gathers (L2-resident z) and coalesced f32 atomics.
// ---------------------------------------------------------------------------
__global__ void edge_aggregate(const int* __restrict__ src,
                               const int* __restrict__ dst,
                               const float* __restrict__ ebuf,
                               const float* __restrict__ s,
                               const float* __restrict__ Z,
                               float* __restrict__ out,
                               int H, int D) {
  int e = blockIdx.x * (blockDim.x >> 5) + (threadIdx.x >> 5);
  if (e >= N_EDGES) return;
  const int lane = threadIdx.x & 31;
  const int si = src[e], di = dst[e];
  const int F = H * D;
  // Pull the gathered source-feature row toward the WGP (global_prefetch_b8).
  __builtin_prefetch(&Z[(size_t)si * F + lane], 0, 0);
  for (int f = lane; f < F; f += 32) {
    int h = f / D;
    float alpha = ebuf[(size_t)e * H + h] / s[(size_t)di * H + h];
    atomicAdd(&out[(size_t)di * F + f], alpha * Z[(size_t)si * F + f]);
  }
}

// ---------------------------------------------------------------------------
// ELU in place (layer-0 activation)
// ---------------------------------------------------------------------------
__global__ void elu_inplace(float* __restrict__ x, int count) {
  int i = blockIdx.x * blockDim.x + threadIdx.x;
  if (i < count) {
    float v = x[i];
    x[i] = v > 0.f ? v : (__expf(v) - 1.f);
  }
}

// ---------------------------------------------------------------------------
// Host-side orchestration
// ---------------------------------------------------------------------------
extern "C" void kernel_launch(void* const* d_in, const int* in_sizes, int n_in,
                              void* d_out, int out_size, void* d_ws, size_t ws_size,
                              hipStream_t stream) {
  (void)in_sizes; (void)n_in; (void)out_size; (void)ws_size;

  const float* x   = (const float*)d_in[0];
  const int*   src = (const int*)  d_in[1];
  const int*   dst = (const int*)  d_in[2];
  const float* W0  = (const float*)d_in[3];
  const float* al0 = (const float*)d_in[4];
  const float* ar0 = (const float*)d_in[5];
  const float* b0  = (const float*)d_in[6];
  const float* W1  = (const float*)d_in[7];
  const float* al1 = (const float*)d_in[8];
  const float* ar1 = (const float*)d_in[9];
  const float* b1  = (const float*)d_in[10];
  float* out = (float*)d_out;

  // Workspace carve-out (fp32 words)
  float* p = (float*)d_ws;
  float* z0   = p; p += (size_t)N_NODES * F0;    // 50000*256
  float* h    = p; p += (size_t)N_NODES * F0;    // aggregated layer-0 out / ELU'd
  float* z1   = p; p += (size_t)N_NODES * NCLS;  // 50000*40
  float* el0  = p; p += (size_t)N_NODES * H0;
  float* er0  = p; p += (size_t)N_NODES * H0;
  float* s0   = p; p += (size_t)N_NODES * H0;
  unsigned int* m0 = (unsigned int*)p; p += (size_t)N_NODES * H0;
  float* eb0  = p; p += (size_t)N_EDGES * H0;    // per-edge logits/alpha, layer 0
  float* el1  = p; p += (size_t)N_NODES;
  float* er1  = p; p += (size_t)N_NODES;
  float* s1   = p; p += (size_t)N_NODES;
  unsigned int* m1 = (unsigned int*)p; p += (size_t)N_NODES;
  float* eb1  = p; p += (size_t)N_EDGES;         // per-edge logits/alpha, layer 1

  const int EH0 = N_EDGES * H0;
  const int NH0 = N_NODES * H0;

  // ---------------- Layer 0 ----------------
  gemm_wmma_f32<<<dim3(N_NODES / 16, 2), 256, 0, stream>>>(x, W0, z0, F0);
  el_er_kernel<<<N_NODES, 32 * H0, 0, stream>>>(z0, al0, ar0, el0, er0, H0, HID);
  init_maxsum<<<(NH0 + 255) / 256, 256, 0, stream>>>(m0, s0, NH0);
  edge_logits<<<(EH0 + 255) / 256, 256, 0, stream>>>(src, dst, el0, er0, eb0, m0, H0);
  edge_exp   <<<(EH0 + 255) / 256, 256, 0, stream>>>(dst, eb0, m0, s0, H0);
  init_out_bias<<<((N_NODES * F0) + 255) / 256, 256, 0, stream>>>(h, b0, N_NODES, F0);
  edge_aggregate<<<(N_EDGES + 7) / 8, 256, 0, stream>>>(src, dst, eb0, s0, z0, h, H0, HID);
  elu_inplace<<<((N_NODES * F0) + 255) / 256, 256, 0, stream>>>(h, N_NODES * F0);

  // ---------------- Layer 1 ----------------
  gemm_wmma_f32<<<dim3(N_NODES / 16, 1), 256, 0, stream>>>(h, W1, z1, NCLS);
  el_er_kernel<<<N_NODES, 32, 0, stream>>>(z1, al1, ar1, el1, er1, 1, NCLS);
  init_maxsum<<<(N_NODES + 255) / 256, 256, 0, stream>>>(m1, s1, N_NODES);
  edge_logits<<<(N_EDGES + 255) / 256, 256, 0, stream>>>(src, dst, el1, er1, eb1, m1, 1);
  edge_exp   <<<(N_EDGES + 255) / 256, 256, 0, stream>>>(dst, eb1, m1, s1, 1);
  // out = bias1 (head mean over H1=1 is identity), then scatter-add messages.
  init_out_bias<<<((N_NODES * NCLS) + 255) / 256, 256, 0, stream>>>(out, b1, N_NODES, NCLS);
  edge_aggregate<<<(N_EDGES + 7) / 8, 256, 0, stream>>>(src, dst, eb1, s1, z1, out, 1, NCLS);
}